// SG_65833258713756
// MI455X (gfx1250) — compile-verified
//
#include <hip/hip_runtime.h>
#include <hip/hip_bf16.h>
#include <math.h>

typedef float v2f __attribute__((ext_vector_type(2)));
typedef float v8f __attribute__((ext_vector_type(8)));

#define NEG_INF (-3.4e38f)

__device__ __forceinline__ v8f wmma4(v2f a, v2f b, v8f c) {
    // D = A(16x4,f32) * B(4x16,f32) + C(16x16,f32)
    return __builtin_amdgcn_wmma_f32_16x16x4_f32(
        /*neg_a=*/false, a, /*neg_b=*/false, b,
        /*c_mod=*/(short)0, c, /*reuse_a=*/false, /*reuse_b=*/false);
}

// ---------------------------------------------------------------------------
// squared norms: sq[i] = sum_c x[i][c]^2
// ---------------------------------------------------------------------------
template <int CIN>
__global__ void sqnorm_kernel(const float* __restrict__ x, float* __restrict__ sq, int total) {
    int i = blockIdx.x * 256 + threadIdx.x;
    if (i >= total) return;
    const float* p = x + (size_t)i * CIN;
    float s = 0.f;
#pragma unroll
    for (int c = 0; c < CIN; ++c) s += p[c] * p[c];
    sq[i] = s;
}

// ---------------------------------------------------------------------------
// KNN: for each row, select indices of the 20 largest values of
//   2*x_i.x_j - ||x_j||^2   (row-constant ||x_i||^2 dropped: order invariant)
// One workgroup = 16 rows, 4 waves; each wave streams 32 column tiles via
// v_wmma_f32_16x16x4_f32 and maintains per-row top-20 lists; merged at end.
// ---------------------------------------------------------------------------
template <int CIN>
__global__ void __launch_bounds__(128)
knn_kernel(const float* __restrict__ x, const float* __restrict__ sq,
           int* __restrict__ idx, int N) {
    constexpr int KST = CIN / 4;
    __shared__ float sqc[2048];
    __shared__ float stage[4][16][16];
    __shared__ float cval[4][16][20];
    __shared__ int   cidx[4][16][20];

    const int tid  = threadIdx.x;
    const int wave = tid >> 5;
    const int lane = tid & 31;
    const int b = blockIdx.y;
    const int rowbase = blockIdx.x * 16;
    const int n16 = lane & 15;
    const int khalf = lane >> 4;

    for (int i = tid; i < N; i += 128) sqc[i] = sq[b * N + i];
    for (int i = tid; i < 4 * 16 * 20; i += 128) {
        (&cval[0][0][0])[i] = NEG_INF;
        (&cidx[0][0][0])[i] = 0;
    }
    __syncthreads();

    // A operand in registers: rows = 16 points of this tile
    v2f a[KST];
    {
        const float* xr = x + ((size_t)b * N + rowbase + n16) * CIN;
#pragma unroll
        for (int kk = 0; kk < KST; ++kk) {
            int k0 = kk * 4 + 2 * khalf;
            a[kk].x = xr[k0];
            a[kk].y = xr[k0 + 1];
        }
    }

    float minv = NEG_INF;  // per-owner-lane running min of its 20-list
    int   minp = 0;

    const int ntile = N / 16;
    for (int t = wave; t < ntile; t += 4) {
        const float* xc = x + ((size_t)b * N + t * 16 + n16) * CIN;
        v8f acc = {};
#pragma unroll
        for (int kk = 0; kk < KST; ++kk) {
            int k0 = kk * 4 + 2 * khalf;
            v2f bv;
            bv.x = xc[k0];
            bv.y = xc[k0 + 1];
            acc = wmma4(a[kk], bv, acc);
        }
        float sc = sqc[t * 16 + n16];
#pragma unroll
        for (int r = 0; r < 8; ++r)
            stage[wave][r + 8 * khalf][n16] = 2.0f * acc[r] - sc;
        __syncthreads();

        if (lane < 16) {
            const int row = lane;
            for (int j = 0; j < 16; ++j) {
                float v = stage[wave][row][j];
                if (v > minv) {
                    cval[wave][row][minp] = v;
                    cidx[wave][row][minp] = t * 16 + j;
                    float mv = cval[wave][row][0];
                    int   mp = 0;
#pragma unroll 5
                    for (int p = 1; p < 20; ++p) {
                        float c = cval[wave][row][p];
                        if (c < mv) { mv = c; mp = p; }
                    }
                    minv = mv;
                    minp = mp;
                }
            }
        }
        __syncthreads();
    }

    // merge 4 partial top-20 lists per row -> final top-20 indices
    if (tid < 16) {
        const int row = tid;
        for (int s = 0; s < 20; ++s) {
            float best = NEG_INF;
            int bw = 0, bp = 0;
            for (int w = 0; w < 4; ++w)
                for (int p = 0; p < 20; ++p) {
                    float v = cval[w][row][p];
                    if (v > best) { best = v; bw = w; bp = p; }
                }
            idx[((size_t)b * N + rowbase + row) * 20 + s] = cidx[bw][row][bp];
            cval[bw][row][bp] = NEG_INF;
        }
    }
}

// ---------------------------------------------------------------------------
// EdgeConv: y[n,o] = max_k LeakyReLU( BN( nbr·Wa^T + ctr·(Wb-Wa)^T ) )
// base = ctr·(Wb-Wa)^T computed once per point as the WMMA C accumulator.
// One workgroup = 16 points, 4 waves; wave w owns output channels [16w,16w+16).
// ---------------------------------------------------------------------------
template <int CIN>
__global__ void __launch_bounds__(128)
edgeconv_kernel(const float* __restrict__ xin, const int* __restrict__ idx,
                const float* __restrict__ W, const float* __restrict__ gam,
                const float* __restrict__ bet, const float* __restrict__ mea,
                const float* __restrict__ var, float* __restrict__ xout, int N) {
    constexpr int KST = CIN / 4;
    __shared__ float wa[CIN][64];   // Wa^T   : wa[c][o] = W[o][c]
    __shared__ float wd[CIN][64];   // (Wb-Wa)^T
    __shared__ float ctr[16][CIN];
    __shared__ float nbr[16][CIN];
    __shared__ float scl[64], sft[64];

    const int tid  = threadIdx.x;
    const int wave = tid >> 5;
    const int lane = tid & 31;
    const int b = blockIdx.y;
    const int rowbase = blockIdx.x * 16;
    const int n16 = lane & 15;
    const int khalf = lane >> 4;
    const int o = wave * 16 + n16;  // this lane's output channel

    for (int i = tid; i < CIN * 64; i += 128) {
        int c = i >> 6, oo = i & 63;
        float w0 = W[oo * (2 * CIN) + c];
        wa[c][oo] = w0;
        wd[c][oo] = W[oo * (2 * CIN) + CIN + c] - w0;
    }
    if (tid < 64) {
        float s = gam[tid] * rsqrtf(var[tid] + 1e-5f);
        scl[tid] = s;
        sft[tid] = bet[tid] - mea[tid] * s;
    }
    for (int i = tid; i < 16 * CIN; i += 128) {
        int m = i / CIN, c = i % CIN;
        ctr[m][c] = xin[((size_t)b * N + rowbase + m) * CIN + c];
    }
    __syncthreads();

    // base accumulator: ctr · (Wb-Wa)^T
    v8f base = {};
#pragma unroll
    for (int kk = 0; kk < KST; ++kk) {
        int k0 = kk * 4 + 2 * khalf;
        v2f av, bv;
        av.x = ctr[n16][k0];
        av.y = ctr[n16][k0 + 1];
        bv.x = wd[k0][o];
        bv.y = wd[k0 + 1][o];
        base = wmma4(av, bv, base);
    }

    const float s  = scl[o];
    const float sh = sft[o];
    v8f vmax;
#pragma unroll
    for (int r = 0; r < 8; ++r) vmax[r] = NEG_INF;

    for (int kk2 = 0; kk2 < 20; ++kk2) {
        for (int i = tid; i < 16 * CIN; i += 128) {
            int m = i / CIN, c = i % CIN;
            int nb = idx[((size_t)b * N + rowbase + m) * 20 + kk2];
            nbr[m][c] = xin[((size_t)b * N + nb) * CIN + c];
        }
        __syncthreads();

        v8f y = base;
#pragma unroll
        for (int kk = 0; kk < KST; ++kk) {
            int k0 = kk * 4 + 2 * khalf;
            v2f av, bv;
            av.x = nbr[n16][k0];
            av.y = nbr[n16][k0 + 1];
            bv.x = wa[k0][o];
            bv.y = wa[k0 + 1][o];
            y = wmma4(av, bv, y);
        }
#pragma unroll
        for (int r = 0; r < 8; ++r) {
            float t = y[r] * s + sh;
            t = (t >= 0.f) ? t : 0.2f * t;
            vmax[r] = fmaxf(vmax[r], t);
        }
        __syncthreads();
    }

#pragma unroll
    for (int r = 0; r < 8; ++r)
        xout[((size_t)b * N + rowbase + r + 8 * khalf) * 64 + o] = vmax[r];
}

// ---------------------------------------------------------------------------
// meanemb[b][f] = mean_n emb[b][n][f]
// ---------------------------------------------------------------------------
__global__ void __launch_bounds__(256)
colmean_kernel(const float* __restrict__ emb, float* __restrict__ meanemb, int N) {
    __shared__ float red[256];
    const int b = blockIdx.x;
    const int f = threadIdx.x & 63, part = threadIdx.x >> 6;
    float s = 0.f;
    for (int n = part; n < N; n += 4) s += emb[((size_t)b * N + n) * 64 + f];
    red[threadIdx.x] = s;
    __syncthreads();
    if (part == 0)
        meanemb[b * 64 + f] = (red[f] + red[64 + f] + red[128 + f] + red[192 + f]) / (float)N;
}

// ctx[b][g] = tanh( meanemb[b] @ W_att[:,g] )
__global__ void ctx_kernel(const float* __restrict__ meanemb,
                           const float* __restrict__ Watt, float* __restrict__ ctx) {
    const int b = blockIdx.x, g = threadIdx.x;  // 64 threads
    float s = 0.f;
    for (int f = 0; f < 64; ++f) s += meanemb[b * 64 + f] * Watt[f * 64 + g];
    ctx[b * 64 + g] = tanhf(s);
}

// e[b][f] = sum_n emb[b][n][f] * sigmoid(emb[b][n] . ctx[b])
__global__ void __launch_bounds__(256)
pool_kernel(const float* __restrict__ emb, const float* __restrict__ ctx,
            float* __restrict__ e, int N) {
    __shared__ float ctxl[64];
    __shared__ float sl[2048];
    __shared__ float red[256];
    const int b = blockIdx.x, tid = threadIdx.x;
    if (tid < 64) ctxl[tid] = ctx[b * 64 + tid];
    __syncthreads();
    for (int n = tid; n < N; n += 256) {
        const float* er = emb + ((size_t)b * N + n) * 64;
        float d = 0.f;
        for (int f = 0; f < 64; ++f) d += er[f] * ctxl[f];
        sl[n] = 1.f / (1.f + expf(-d));
    }
    __syncthreads();
    const int f = tid & 63, part = tid >> 6;
    float s = 0.f;
    for (int n = part; n < N; n += 4) s += emb[((size_t)b * N + n) * 64 + f] * sl[n];
    red[tid] = s;
    __syncthreads();
    if (part == 0) e[b * 64 + f] = red[f] + red[64 + f] + red[128 + f] + red[192 + f];
}

// out[b][t] = relu( e1 @ W_t[:,:,t] @ e2 + W_block[t] . [e1;e2] + bias[t] )
__global__ void __launch_bounds__(128)
tn_kernel(const float* __restrict__ e1, const float* __restrict__ e2,
          const float* __restrict__ Wt, const float* __restrict__ Wb,
          const float* __restrict__ bias, float* __restrict__ out) {
    __shared__ float red[128];
    __shared__ float e1l[64], e2l[64];
    const int b = blockIdx.x, tid = threadIdx.x;
    const int t = tid & 15, part = tid >> 4;  // 8 parts
    if (tid < 64) { e1l[tid] = e1[b * 64 + tid]; e2l[tid] = e2[b * 64 + tid]; }
    __syncthreads();
    float s = 0.f;
    for (int f = part; f < 64; f += 8) {
        float inner = 0.f;
        for (int g = 0; g < 64; ++g) inner += Wt[((size_t)(f * 64 + g)) * 16 + t] * e2l[g];
        s += e1l[f] * inner;
    }
    red[tid] = s;
    __syncthreads();
    if (part == 0) {
        float tot = 0.f;
        for (int p = 0; p < 8; ++p) tot += red[p * 16 + t];
        float blk = 0.f;
        for (int f = 0; f < 64; ++f)
            blk += Wb[t * 128 + f] * e1l[f] + Wb[t * 128 + 64 + f] * e2l[f];
        float r = tot + blk + bias[t];
        out[b * 16 + t] = (r > 0.f) ? r : 0.f;
    }
}

// ---------------------------------------------------------------------------
extern "C" void kernel_launch(void* const* d_in, const int* in_sizes, int n_in,
                              void* d_out, int out_size, void* d_ws, size_t ws_size,
                              hipStream_t stream) {
    (void)in_sizes; (void)n_in; (void)out_size; (void)ws_size;
    const int B = 8, N = 2048, F = 64;

    const float* graph1  = (const float*)d_in[0];
    const float* graph2  = (const float*)d_in[1];
    const float* W1 = (const float*)d_in[2];
    const float* g1 = (const float*)d_in[3];
    const float* b1 = (const float*)d_in[4];
    const float* m1 = (const float*)d_in[5];
    const float* v1 = (const float*)d_in[6];
    const float* W2 = (const float*)d_in[7];
    const float* g2 = (const float*)d_in[8];
    const float* b2 = (const float*)d_in[9];
    const float* m2 = (const float*)d_in[10];
    const float* v2 = (const float*)d_in[11];
    const float* W3 = (const float*)d_in[12];
    const float* g3 = (const float*)d_in[13];
    const float* b3 = (const float*)d_in[14];
    const float* m3 = (const float*)d_in[15];
    const float* v3 = (const float*)d_in[16];
    const float* Watt = (const float*)d_in[17];
    const float* Wt   = (const float*)d_in[18];
    const float* Wblk = (const float*)d_in[19];
    const float* btn  = (const float*)d_in[20];

    float* ws = (float*)d_ws;
    const size_t BNF = (size_t)B * N * F;
    float* bufA    = ws;
    float* bufB    = bufA + BNF;
    float* sqbuf   = bufB + BNF;
    int*   idxbuf  = (int*)(sqbuf + (size_t)B * N);
    float* meanemb = (float*)(idxbuf + (size_t)B * N * 20);
    float* ctxbuf  = meanemb + B * F;
    float* e1vec   = ctxbuf + B * F;
    float* e2vec   = e1vec + B * F;

    dim3 tilegrid(N / 16, B);
    const int sqblocks = (B * N + 255) / 256;

    auto run_graph = [&](const float* xin, float* evec) {
        // layer 1 (C=12)
        sqnorm_kernel<12><<<sqblocks, 256, 0, stream>>>(xin, sqbuf, B * N);
        knn_kernel<12><<<tilegrid, 128, 0, stream>>>(xin, sqbuf, idxbuf, N);
        edgeconv_kernel<12><<<tilegrid, 128, 0, stream>>>(xin, idxbuf, W1, g1, b1, m1, v1, bufA, N);
        // layer 2 (C=64)
        sqnorm_kernel<64><<<sqblocks, 256, 0, stream>>>(bufA, sqbuf, B * N);
        knn_kernel<64><<<tilegrid, 128, 0, stream>>>(bufA, sqbuf, idxbuf, N);
        edgeconv_kernel<64><<<tilegrid, 128, 0, stream>>>(bufA, idxbuf, W2, g2, b2, m2, v2, bufB, N);
        // layer 3 (C=64)
        sqnorm_kernel<64><<<sqblocks, 256, 0, stream>>>(bufB, sqbuf, B * N);
        knn_kernel<64><<<tilegrid, 128, 0, stream>>>(bufB, sqbuf, idxbuf, N);
        edgeconv_kernel<64><<<tilegrid, 128, 0, stream>>>(bufB, idxbuf, W3, g3, b3, m3, v3, bufA, N);
        // attention pooling
        colmean_kernel<<<B, 256, 0, stream>>>(bufA, meanemb, N);
        ctx_kernel<<<B, 64, 0, stream>>>(meanemb, Watt, ctxbuf);
        pool_kernel<<<B, 256, 0, stream>>>(bufA, ctxbuf, evec, N);
    };

    run_graph(graph1, e1vec);
    run_graph(graph2, e2vec);
    tn_kernel<<<B, 128, 0, stream>>>(e1vec, e2vec, Wt, Wblk, btn, (float*)d_out);
}